// GATNet_34479997452470
// MI455X (gfx1250) — compile-verified
//
#include <hip/hip_runtime.h>
#include <math.h>

// ---------------------------------------------------------------------------
// GAT 2-layer forward for MI455X (gfx1250, wave32).
// GEMMs use V_WMMA_F32_16X16X4_F32 (exact fp32). Edge stage is one wave per
// edge with wave32 butterfly reduction and ordered-uint atomicMax scatter.
// ---------------------------------------------------------------------------

typedef __attribute__((ext_vector_type(2))) float v2f;
typedef __attribute__((ext_vector_type(8))) float v8f;

#define CDIM 32   // CIN == COUT == 32
#define BDIM 2    // batch

// Order-preserving float <-> uint encoding (monotone under unsigned compare)
__device__ __forceinline__ unsigned enc_ord(float f) {
  unsigned u = __float_as_uint(f);
  return (u & 0x80000000u) ? ~u : (u | 0x80000000u);
}
__device__ __forceinline__ float dec_ord(unsigned u) {
  return (u & 0x80000000u) ? __uint_as_float(u & 0x7fffffffu)
                           : __uint_as_float(~u);
}
#define AGGR_INIT 0x007fffffu  // enc_ord(-inf)

// ---------------------------------------------------------------------------
// out[M,32] = in[M,32] @ W[32,32]   (row-major), exact fp32 via WMMA 16x16x4.
// One wave per 16 output rows; both 16-col tiles; K=32 as 8 chained WMMAs.
// ---------------------------------------------------------------------------
__global__ __launch_bounds__(256) void gat_gemm32(
    const float* __restrict__ in, const float* __restrict__ W,
    float* __restrict__ out, int M) {
  int wave = (int)((blockIdx.x * blockDim.x + threadIdx.x) >> 5);
  int lane = (int)(threadIdx.x & 31);
  int rowbase = wave << 4;
  if (rowbase >= M) return;  // wave-uniform: EXEC stays all-ones for WMMA

  int r  = lane & 15;        // row (for A) / col (for B,C)
  int kh = (lane >> 4) << 1; // K sub-pair select (0 or 2)

  v8f acc0 = {};  // cols 0..15
  v8f acc1 = {};  // cols 16..31
#pragma unroll
  for (int kb = 0; kb < CDIM; kb += 4) {
    int rr = rowbase + r;
    if (rr >= M) rr = M - 1;  // clamp tail rows (stores are masked later)
    const float* ip = in + rr * CDIM + kb + kh;
    v2f a, b0, b1;
    a.x = ip[0];
    a.y = ip[1];
    const float* wp = W + (kb + kh) * CDIM;  // row (kb+kh) of W
    b0.x = wp[r];
    b0.y = wp[CDIM + r];
    b1.x = wp[16 + r];
    b1.y = wp[CDIM + 16 + r];
    acc0 = __builtin_amdgcn_wmma_f32_16x16x4_f32(false, a, false, b0,
                                                 (short)0, acc0, false, false);
    acc1 = __builtin_amdgcn_wmma_f32_16x16x4_f32(false, a, false, b1,
                                                 (short)0, acc1, false, false);
  }

  int rh = (lane >> 4) << 3;  // lanes 16-31 hold rows rowbase+8..rowbase+15
#pragma unroll
  for (int v = 0; v < 8; ++v) {
    int row = rowbase + rh + v;
    if (row < M) {
      out[row * CDIM + r]      = acc0[v];
      out[row * CDIM + 16 + r] = acc1[v];
    }
  }
}

// ---------------------------------------------------------------------------
// Fill aggregation buffer with encoded -inf.
// ---------------------------------------------------------------------------
__global__ __launch_bounds__(256) void gat_fill(unsigned* __restrict__ p, int n) {
  int i = (int)(blockIdx.x * blockDim.x + threadIdx.x);
  if (i < n) p[i] = AGGR_INIT;
}

// ---------------------------------------------------------------------------
// One wave per edge: gather xu[src], xu[dst] (coalesced 128B rows), wave32
// butterfly dot -> sigmoid gate, scatter-max encoded message into aggr[dst].
// ---------------------------------------------------------------------------
__global__ __launch_bounds__(256) void gat_edge(
    const int* __restrict__ ei, const float* __restrict__ enorm,
    const float* __restrict__ xu, unsigned* __restrict__ aggr, int E) {
  int wid  = (int)((blockIdx.x * blockDim.x + threadIdx.x) >> 5);
  int lane = (int)(threadIdx.x & 31);
  if (wid >= E) return;

  int   src = ei[wid];      // edge_index[0][e]  -> x_j
  int   dst = ei[E + wid];  // edge_index[1][e]  -> x_i (aggregation target)
  float nrm = enorm[wid];

#pragma unroll
  for (int b = 0; b < BDIM; ++b) {
    float xj = xu[(src * BDIM + b) * CDIM + lane];
    float xi = xu[(dst * BDIM + b) * CDIM + lane];
    float s  = xi * xj;
#pragma unroll
    for (int o = 16; o > 0; o >>= 1) s += __shfl_xor(s, o, 32);
    float gate = 1.0f / (1.0f + __expf(-s));
    float msg  = xj * gate * nrm;
    atomicMax(&aggr[(dst * BDIM + b) * CDIM + lane], enc_ord(msg));
  }
}

// ---------------------------------------------------------------------------
// out = leaky_relu(xn + (aggr finite ? aggr : 0), 0.01)
// ---------------------------------------------------------------------------
__global__ __launch_bounds__(256) void gat_final(
    const float* __restrict__ xn, const unsigned* __restrict__ aggr,
    float* __restrict__ out, int n) {
  int i = (int)(blockIdx.x * blockDim.x + threadIdx.x);
  if (i >= n) return;
  unsigned u = aggr[i];
  float a = (u == AGGR_INIT) ? 0.0f : dec_ord(u);  // empty segment -> 0
  float v = xn[i] + a;
  out[i] = v > 0.0f ? v : 0.01f * v;
}

// ---------------------------------------------------------------------------
// Host-side orchestration. ws layout: 3 regions of M*32 floats, ping-ponged.
// ---------------------------------------------------------------------------
extern "C" void kernel_launch(void* const* d_in, const int* in_sizes, int n_in,
                              void* d_out, int out_size, void* d_ws,
                              size_t ws_size, hipStream_t stream) {
  (void)n_in; (void)out_size; (void)ws_size;

  const float* X     = (const float*)d_in[0];
  const int*   ei    = (const int*)d_in[1];
  // d_in[2] edge_attr, d_in[5] We1, d_in[8] We2: unused in reference message
  const float* enorm = (const float*)d_in[3];
  const float* Wn1   = (const float*)d_in[4];
  const float* U1    = (const float*)d_in[6];
  const float* Wn2   = (const float*)d_in[7];
  const float* U2    = (const float*)d_in[9];

  const int N = in_sizes[0] / (BDIM * CDIM);
  const int E = in_sizes[1] / 2;
  const int M = N * BDIM;          // 100000 rows of 32 channels
  const int elems = M * CDIM;      // 3.2M floats per buffer

  float* R0 = (float*)d_ws;        // xn
  float* R1 = R0 + elems;          // xu / c1 / aggr2
  float* R2 = R1 + elems;          // aggr1 / xu2
  float* out = (float*)d_out;

  dim3 blk(256);
  const int gemmBlocks = (((M + 15) / 16) + 7) / 8;  // 8 waves (16 rows each)/block
  const int fillBlocks = (elems + 255) / 256;
  const int edgeBlocks = (E + 7) / 8;                // 8 edges (waves)/block

  // ---- layer 1: c1 = leaky_relu(gat_conv(X, Wn1, U1)) ----
  gat_gemm32<<<gemmBlocks, blk, 0, stream>>>(X, Wn1, R0, M);   // xn = X@Wn1
  gat_gemm32<<<gemmBlocks, blk, 0, stream>>>(R0, U1, R1, M);   // xu = xn@U1
  gat_fill  <<<fillBlocks, blk, 0, stream>>>((unsigned*)R2, elems);
  gat_edge  <<<edgeBlocks, blk, 0, stream>>>(ei, enorm, R1, (unsigned*)R2, E);
  gat_final <<<fillBlocks, blk, 0, stream>>>(R0, (const unsigned*)R2, R1, elems); // c1 -> R1

  // ---- layer 2: out = leaky_relu(gat_conv(c1, Wn2, U2)) ----
  gat_gemm32<<<gemmBlocks, blk, 0, stream>>>(R1, Wn2, R0, M);  // xn = c1@Wn2
  gat_gemm32<<<gemmBlocks, blk, 0, stream>>>(R0, U2, R2, M);   // xu = xn@U2
  gat_fill  <<<fillBlocks, blk, 0, stream>>>((unsigned*)R1, elems);
  gat_edge  <<<edgeBlocks, blk, 0, stream>>>(ei, enorm, R2, (unsigned*)R1, E);
  gat_final <<<fillBlocks, blk, 0, stream>>>(R0, (const unsigned*)R1, out, elems);
}